// Analogy_RE_Model_89756226552375
// MI455X (gfx1250) — compile-verified
//
#include <hip/hip_runtime.h>
#include <math.h>

typedef __attribute__((ext_vector_type(2))) float v2f;
typedef __attribute__((ext_vector_type(8))) float v8f;

#define DIM  512
#define NPOS 512
#define NNEG 512
#define NALL 1024

// ---------------------------------------------------------------- zero init
__global__ void zero_ws(float* __restrict__ deno, float* __restrict__ out) {
  int i = blockIdx.x * blockDim.x + threadIdx.x;
  if (i < NPOS) deno[i] = 0.0f;
  if (i == 0) out[0] = 0.0f;
}

// ------------------------------------------------- row normalize + t1 / t2
// One 256-thread block per row r in [0, 1024). Fuses:
//   posn/negn = row / max(||row||, 1e-8)
//   t2[r] = row . w2 ;  t1[r] = row . w1 (pos rows only)
__global__ void rownorm(const float* __restrict__ pos, const float* __restrict__ neg,
                        const float* __restrict__ w,
                        float* __restrict__ posn, float* __restrict__ negn,
                        float* __restrict__ t1, float* __restrict__ t2) {
  int r = blockIdx.x;
  int t = threadIdx.x;  // 256 threads, 2 elements each
  const float* src = (r < NPOS) ? (pos + r * DIM) : (neg + (r - NPOS) * DIM);
  float*       dst = (r < NPOS) ? (posn + r * DIM) : (negn + (r - NPOS) * DIM);

  float v0 = src[t], v1 = src[t + 256];
  float ss = v0 * v0 + v1 * v1;
  float d2 = v0 * w[DIM + t] + v1 * w[DIM + t + 256];
  float d1 = (r < NPOS) ? (v0 * w[t] + v1 * w[t + 256]) : 0.0f;

  __shared__ float red[3][256];
  red[0][t] = ss; red[1][t] = d1; red[2][t] = d2;
  __syncthreads();
  for (int s = 128; s > 0; s >>= 1) {
    if (t < s) {
      red[0][t] += red[0][t + s];
      red[1][t] += red[1][t + s];
      red[2][t] += red[2][t + s];
    }
    __syncthreads();
  }
  float inv = 1.0f / fmaxf(sqrtf(red[0][0]), 1e-8f);
  dst[t] = v0 * inv;
  dst[t + 256] = v1 * inv;
  if (t == 0) {
    t2[r] = red[2][0];
    if (r < NPOS) t1[r] = red[1][0];
  }
}

// --------------------------------------------- WMMA f32 16x16x4 tile GEMM
// Computes C(16x16) = Arows(tile tm) . Brows(tile tn)^T over K = DIM.
// A-layout (16x4): lane(0-15) holds M=lane, K = 2*(lane>>4)+{v0,v1}
// B-layout (4x16): symmetric -> both operands load float2 from their row.
__device__ __forceinline__ v8f wmma_dot_tile(const float* __restrict__ A,
                                             const float* __restrict__ B,
                                             int tm, int tn) {
  int lane = threadIdx.x & 31;
  int half = lane >> 4;
  int l16  = lane & 15;
  const float* ar = A + (tm * 16 + l16) * DIM;
  const float* br = B + (tn * 16 + l16) * DIM;
  v8f c = {0.f, 0.f, 0.f, 0.f, 0.f, 0.f, 0.f, 0.f};
  for (int k = 0; k < DIM; k += 4) {
    int kk = k + 2 * half;
    v2f a, b;
    a.x = ar[kk]; a.y = ar[kk + 1];
    b.x = br[kk]; b.y = br[kk + 1];
    c = __builtin_amdgcn_wmma_f32_16x16x4_f32(false, a, false, b,
                                              (short)0, c, false, false);
  }
  return c;
}

// exp(S_pn) row-sums -> deno (atomic accumulate across n-tiles)
__global__ void cos_pn_deno(const float* __restrict__ posn,
                            const float* __restrict__ negn,
                            float* __restrict__ deno) {
  int tm = blockIdx.x, tn = blockIdx.y;
  int lane = threadIdx.x;
  v8f c = wmma_dot_tile(posn, negn, tm, tn);
  int half = lane >> 4, l16 = lane & 15;
  // D layout: c[v] = S[tm*16 + v + 8*half][tn*16 + l16]
  for (int v = 0; v < 8; v++) {
    float e = expf(c[v]);
    for (int off = 1; off < 16; off <<= 1) e += __shfl_xor(e, off, 32);
    if (l16 == 0) atomicAdd(&deno[tm * 16 + v + 8 * half], e);
  }
}

// sum over (i,j) of log(deno[i] + exp(s) + eps) - s   for S_pp = posn.posn^T
__global__ void cos_pp_loss(const float* __restrict__ posn,
                            const float* __restrict__ deno,
                            float* __restrict__ out) {
  int tm = blockIdx.x, tn = blockIdx.y;
  int lane = threadIdx.x;
  v8f c = wmma_dot_tile(posn, posn, tm, tn);
  int half = lane >> 4;
  float acc = 0.0f;
  for (int v = 0; v < 8; v++) {
    int m = tm * 16 + v + 8 * half;
    float s = c[v];
    acc += logf(deno[m] + expf(s) + 1e-5f) - s;
  }
  for (int off = 16; off; off >>= 1) acc += __shfl_xor(acc, off, 32);
  if (lane == 0) atomicAdd(out, acc);
}

// --------------------------------------------------------------- BCE part
// t3[i][j] = sum_d |pos[i][d]-allv[j][d]| * w3[d]; logits -> softplus BCE.
#define BK 128
__global__ void bce_loss(const float* __restrict__ pos, const float* __restrict__ neg,
                         const float* __restrict__ w, const float* __restrict__ bias,
                         const float* __restrict__ t1, const float* __restrict__ t2,
                         float* __restrict__ out) {
  int i0 = blockIdx.x * 16;  // pos-row tile
  int j0 = blockIdx.y * 16;  // allv-row tile
  int t  = threadIdx.x;      // 256 threads
  int tx = t & 15, ty = t >> 4;

  __shared__ float sw[DIM];
  __shared__ float sp[16 * (BK + 1)];  // +1 pad: avoid 16-way bank conflict
  __shared__ float sa[16 * (BK + 1)];
  for (int c = t; c < DIM; c += 256) sw[c] = w[2 * DIM + c];

  float t3 = 0.0f;
  for (int kc = 0; kc < DIM; kc += BK) {
    __syncthreads();
    for (int idx = t; idx < 16 * BK; idx += 256) {
      int row = idx / BK, col = idx % BK;
      sp[row * (BK + 1) + col] = pos[(i0 + row) * DIM + kc + col];
      int jr = j0 + row;
      sa[row * (BK + 1) + col] = (jr < NPOS) ? pos[jr * DIM + kc + col]
                                             : neg[(jr - NPOS) * DIM + kc + col];
    }
    __syncthreads();
    const float* pr = sp + ty * (BK + 1);
    const float* ar = sa + tx * (BK + 1);
    for (int dd = 0; dd < BK; dd++)
      t3 += fabsf(pr[dd] - ar[dd]) * sw[kc + dd];
  }

  float logit = t1[i0 + ty] + t2[j0 + tx] + t3 + bias[0];
  float lse   = log1pf(expf(-fabsf(logit)));
  float bce   = (j0 + tx < NPOS) ? (fmaxf(-logit, 0.0f) + lse)   // softplus(-x)
                                 : (fmaxf(logit, 0.0f) + lse);   // softplus(x)
  float acc = bce * (1.0f / (float)NALL);

  __shared__ float red[256];
  red[t] = acc;
  __syncthreads();
  for (int s = 128; s > 0; s >>= 1) {
    if (t < s) red[t] += red[t + s];
    __syncthreads();
  }
  if (t == 0) atomicAdd(out, red[0]);
}

// ---------------------------------------------------------------- launcher
extern "C" void kernel_launch(void* const* d_in, const int* in_sizes, int n_in,
                              void* d_out, int out_size, void* d_ws, size_t ws_size,
                              hipStream_t stream) {
  const float* pos  = (const float*)d_in[0];  // 512x512
  const float* neg  = (const float*)d_in[1];  // 512x512
  const float* w    = (const float*)d_in[2];  // 1x1536
  const float* bias = (const float*)d_in[3];  // 1
  float* out = (float*)d_out;

  float* ws   = (float*)d_ws;
  float* posn = ws;                      // 512*512
  float* negn = posn + NPOS * DIM;       // 512*512
  float* t1   = negn + NNEG * DIM;       // 512
  float* t2   = t1 + NPOS;               // 1024
  float* deno = t2 + NALL;               // 512

  zero_ws<<<2, 256, 0, stream>>>(deno, out);
  rownorm<<<NALL, 256, 0, stream>>>(pos, neg, w, posn, negn, t1, t2);

  dim3 gt(NPOS / 16, NNEG / 16);  // 32x32 tiles, 1 wave each
  cos_pn_deno<<<gt, 32, 0, stream>>>(posn, negn, deno);
  cos_pp_loss<<<gt, 32, 0, stream>>>(posn, deno, out);

  dim3 gb(NPOS / 16, NALL / 16);  // 32x64 tiles, 256 threads each
  bce_loss<<<gb, 256, 0, stream>>>(pos, neg, w, bias, t1, t2, out);
}